// Unresample_58463094833218
// MI455X (gfx1250) — compile-verified
//
#include <hip/hip_runtime.h>
#include <math.h>

typedef __attribute__((ext_vector_type(2))) float v2f;
typedef __attribute__((ext_vector_type(4))) float v4f;
typedef __attribute__((ext_vector_type(8))) float v8f;

namespace {
constexpr int kB    = 4;
constexpr int kC    = 16;
constexpr int kHin  = 512;
constexpr int kWin  = 1024;
constexpr int kHout = 512;
constexpr int kWout = 1024;
constexpr int kHWin  = kHin * kWin;    // input plane elements
constexpr int kHWout = kHout * kWout;  // output plane elements
constexpr int kWavesPerBlock = 8;
constexpr int kPixPerTile    = 16;     // one 16x16 WMMA tile: 16 pixels x 16 channels
}

// 8-byte pair with 4-byte alignment: the (x0, x0+1) horizontal bilinear pair.
struct F2u { float lo, hi; };

__global__ __launch_bounds__(256) void unresample_wmma_f32(
    const float* __restrict__ x,     // [B, C, Hin, Win]
    const float* __restrict__ smap,  // [Hout, Wout, 2] (sx, sy)
    float* __restrict__ out)         // [B, C, Hout, Wout]
{
  const int lane  = threadIdx.x & 31;
  const int wave  = threadIdx.x >> 5;
  const int laneP = lane & 15;   // pixel-in-tile (for A/weights) / channel (for B/D)
  const int hiH   = lane >> 4;   // 0 -> y0 row (K=0,1), 1 -> y1 row (K=2,3)

  const int tile = blockIdx.x * kWavesPerBlock + wave;
  const int P0   = tile * kPixPerTile;  // 16 consecutive output pixels (never straddles a row)

  // ---- Decode this lane's sample (pixel P0+laneP); lanes 16..31 mirror 0..15 ----
  const float sx  = smap[2 * (P0 + laneP) + 0];
  const float sy  = smap[2 * (P0 + laneP) + 1];
  const float x0f = floorf(sx);
  const float y0f = floorf(sy);
  const float dx  = sx - x0f;
  const float dy  = sy - y0f;
  const int   x0  = (int)x0f;           // sx < Win-1 strictly => x1 = x0+1 in bounds
  const int   y0  = (int)y0f;           // sy < Hin-1 strictly => y1 = y0+1 in bounds
  const int   idx00 = y0 * kWin + x0;   // plane-relative element index of (y0, x0)

  // A-matrix 16x4 layout: lane p -> (A[p,0], A[p,1]); lane 16+p -> (A[p,2], A[p,3])
  // K order: 0=(y0,x0) 1=(y0,x1) 2=(y1,x0) 3=(y1,x1)
  const float wy  = hiH ? dy : (1.0f - dy);
  const float wa0 = (1.0f - dx) * wy;
  const float wa1 = dx * wy;

  // B-matrix gather offset: channel = laneP (N), row select = hiH (y0 vs y1).
  // One B64 load fills (B[K,n], B[K+1,n]) = contiguous (x0, x0+1) pair.
  const int voff = (laneP * kHin + hiH) * kWin;  // element offset within batch

#pragma unroll
  for (int b = 0; b < kB; ++b) {
    const float* __restrict__ xb = x + (size_t)b * kC * kHWin;

    v8f accE = {};  // even-q accumulator
    v8f accO = {};  // odd-q accumulator (shortens dependent WMMA chain)

#pragma unroll
    for (int q = 0; q < 16; ++q) {
      // Broadcast pixel q's (y0*Win + x0) to an SGPR -> gathers become SADDR + v_offset B64s
      const int sidx = __builtin_amdgcn_readlane(idx00, q);
      const F2u pv   = *(const F2u*)(xb + sidx + voff);
      v2f bm; bm.x = pv.lo; bm.y = pv.hi;

      // A_q = full weight matrix masked to row q
      const bool sel = (laneP == q);
      v2f am; am.x = sel ? wa0 : 0.0f; am.y = sel ? wa1 : 0.0f;

      if (q & 1)
        accO = __builtin_amdgcn_wmma_f32_16x16x4_f32(
            false, am, false, bm, (short)0, accO, false, false);
      else
        accE = __builtin_amdgcn_wmma_f32_16x16x4_f32(
            false, am, false, bm, (short)0, accE, false, false);
    }

    const v8f acc = accE + accO;

    // D layout: lane (hiH, laneP) holds pixels P0+8*hiH .. P0+8*hiH+7 of channel laneP,
    // i.e. already in store order -> two aligned 16B stores, no shuffle needed.
    float* __restrict__ op =
        out + ((size_t)b * kC + laneP) * kHWout + P0 + 8 * hiH;
    v4f s0 = {acc[0], acc[1], acc[2], acc[3]};
    v4f s1 = {acc[4], acc[5], acc[6], acc[7]};
    *(v4f*)(op + 0) = s0;
    *(v4f*)(op + 4) = s1;
  }
}

extern "C" void kernel_launch(void* const* d_in, const int* in_sizes, int n_in,
                              void* d_out, int out_size, void* d_ws, size_t ws_size,
                              hipStream_t stream) {
  (void)in_sizes; (void)n_in; (void)out_size; (void)d_ws; (void)ws_size;
  const float* x    = (const float*)d_in[0];
  const float* smap = (const float*)d_in[1];
  float* out        = (float*)d_out;

  const int tiles  = kHWout / kPixPerTile;          // 32768
  const int blocks = tiles / kWavesPerBlock;        // 4096, always full -> EXEC all-ones
  hipLaunchKernelGGL(unresample_wmma_f32, dim3(blocks), dim3(256), 0, stream,
                     x, smap, out);
}